// kagconv_53867479826613
// MI455X (gfx1250) — compile-verified
//
#include <hip/hip_runtime.h>
#include <hip/hip_bf16.h>
#include <stddef.h>
#include <stdint.h>

// ---------------------------------------------------------------------------
// Problem constants (from the reference)
// ---------------------------------------------------------------------------
#define BB 32
#define TT 12
#define VV 1024
#define EE 64
#define GG 32
#define BT 384            // B*T
#define ALPHA 0.3f

typedef __attribute__((ext_vector_type(2))) float v2f;
typedef __attribute__((ext_vector_type(8))) float v8f;

// D = A(16x4, f32) * B(4x16, f32) + C(16x16, f32)
__device__ __forceinline__ v8f wmma_f32(v2f a, v2f b, v8f c) {
    return __builtin_amdgcn_wmma_f32_16x16x4_f32(false, a, false, b,
                                                 (short)0, c, false, false);
}

// ---------------------------------------------------------------------------
// CDNA5 async global->LDS copies (ASYNCcnt path, CDNA5 ISA).
// LDS byte address = low 32 bits of the generic pointer.
// saddr = wave-uniform 64-bit base in SGPR pair,
// vaddr = per-lane signed 32-bit byte offset (GVS mode).
// ---------------------------------------------------------------------------
__device__ __forceinline__ uint32_t lds_off(const void* p) {
    return (uint32_t)(uintptr_t)p;
}
__device__ __forceinline__ void async_b32(uint32_t ldso, uint32_t goff,
                                          const void* sbase) {
    asm volatile("global_load_async_to_lds_b32 %0, %1, %2"
                 :: "v"(ldso), "v"(goff), "s"(sbase) : "memory");
}
__device__ __forceinline__ void async_b128(uint32_t ldso, uint32_t goff,
                                           const void* sbase) {
    asm volatile("global_load_async_to_lds_b128 %0, %1, %2"
                 :: "v"(ldso), "v"(goff), "s"(sbase) : "memory");
}
#define ASYNC_WAIT() asm volatile("s_wait_asynccnt 0x0" ::: "memory")

// ---------------------------------------------------------------------------
// Kernel 0: transpose inputs [B,T,V,E] (btge) -> Xt [E][V][T][B]
// Load side: async b128 direct to LDS. Store side: regular (transposed read).
// ---------------------------------------------------------------------------
__global__ __launch_bounds__(256) void k0_transpose(
        const float* __restrict__ in, float* __restrict__ Xt) {
    const int g = blockIdx.x;      // node
    const int t = blockIdx.y;      // timestep
    __shared__ float tile[BB * 65];           // [b][e], pad 65 vs bank conflicts
    const float* ibase = in + ((size_t)t * VV + g) * EE;   // + b*T*V*E + e
    for (int idx = threadIdx.x; idx < BB * (EE / 4); idx += 256) {
        const int b = idx >> 4, q = idx & 15;             // 16 float4 per row
        async_b128(lds_off(&tile[b * 65 + q * 4]),
                   (uint32_t)b * (TT * VV * EE * 4u) + (uint32_t)q * 16u,
                   ibase);
    }
    ASYNC_WAIT();
    __syncthreads();
    for (int idx = threadIdx.x; idx < BB * EE; idx += 256) {
        const int e = idx >> 5, b = idx & 31;
        Xt[(((size_t)e * VV + g) * TT + t) * BB + b] = tile[b * 65 + e];
    }
}

// ---------------------------------------------------------------------------
// Kernel 1: fused relation + graph mixing, per (v_tile of 16, channel e).
//   R_e[v0:v0+16, :] = relu(S_v[16,32] @ S_g[1024,32]^T)   (chunked)
//   Xr_e[v0:v0+16, 0:384] = R_e @ Xt_e[1024, 384]
// Never materializes the 268 MB relation tensor. Grid = (64 v-tiles, 64 e).
// Waves 0-1 build the relation chunk, all 8 waves accumulate 16x384 output.
// All staging goes through the async-to-LDS path.
// ---------------------------------------------------------------------------
#define GCH 32                  // g-chunk
#define SVP 33                  // padded row for S_v / R
#define SGP 33                  // padded row for S_g
#define XSP 388                 // padded row for staged Xt chunk (384 + 4)

__global__ __launch_bounds__(256) void k1_relmix(
        const float* __restrict__ srpe,
        const float* __restrict__ Xt,
        float* __restrict__ Xr) {
    const int v0   = blockIdx.x * 16;
    const int e    = blockIdx.y;
    const int tid  = threadIdx.x;
    const int lane = tid & 31;
    const int wave = tid >> 5;
    const int hl   = lane & 15;              // half-lane (column / row id)
    const int kl   = (lane >> 4) << 1;       // K pair offset: 0 or 2

    __shared__ float sv[16 * SVP];           // S_v tile  [16][32]
    __shared__ float sg[GCH * SGP];          // S_g chunk [32][32]
    __shared__ float rs[16 * SVP];           // relu'd relation chunk [16][32]
    __shared__ float xs[GCH * XSP];          // Xt chunk  [32][384]

    const float* sbase = srpe + e;           // uniform channel base

    // stage S_v once (async b32 gather; srpe is 8MB -> L2 resident)
    for (int idx = tid; idx < 16 * GG; idx += 256) {
        const int m = idx >> 5, k = idx & 31;
        async_b32(lds_off(&sv[m * SVP + k]),
                  (uint32_t)(((v0 + m) * GG + k) * EE) * 4u, sbase);
    }

    v8f acc[3] = {};   // this wave's 3 output tiles over bt

    const float* xbase = Xt + (size_t)e * VV * BT;   // uniform channel base

    for (int g0 = 0; g0 < VV; g0 += GCH) {
        __syncthreads();   // previous chunk fully consumed

        // --- stage S_g chunk (async b32 gather) ---
        for (int idx = tid; idx < GCH * GG; idx += 256) {
            const int i = idx >> 5, k = idx & 31;
            async_b32(lds_off(&sg[i * SGP + k]),
                      (uint32_t)(((g0 + i) * GG + k) * EE) * 4u, sbase);
        }
        // --- stage Xt chunk: contiguous 32*384 floats, async b128 ---
        for (int idx = tid; idx < GCH * (BT / 4); idx += 256) {
            const int r = idx / (BT / 4);
            const int q = idx - r * (BT / 4);
            async_b128(lds_off(&xs[r * XSP + q * 4]),
                       (uint32_t)(g0 * BT) * 4u + (uint32_t)idx * 16u, xbase);
        }
        ASYNC_WAIT();
        __syncthreads();

        // --- step 1: waves 0-1 compute relu(S_v @ S_g^T) -> rs [16][32] ---
        if (wave < 2) {
            const int n0 = wave * 16;
            v8f r = {};
#pragma unroll
            for (int kk = 0; kk < 8; ++kk) {        // K = 32
                v2f a, b;
                a.x = sv[hl * SVP + kk * 4 + kl];
                a.y = sv[hl * SVP + kk * 4 + kl + 1];
                b.x = sg[(n0 + hl) * SGP + kk * 4 + kl];
                b.y = sg[(n0 + hl) * SGP + kk * 4 + kl + 1];
                r = wmma_f32(a, b, r);
            }
#pragma unroll
            for (int i = 0; i < 8; ++i) {
                const int m = i + ((lane >> 4) << 3);
                const float val = r[i] > 0.0f ? r[i] : 0.0f;   // ReLU
                rs[m * SVP + n0 + hl] = val;
            }
        }
        __syncthreads();

        // --- step 2: all waves accumulate Out += R_chunk @ X_chunk ---
#pragma unroll
        for (int kk = 0; kk < 8; ++kk) {            // K = 32 (chunk)
            v2f a;
            a.x = rs[hl * SVP + kk * 4 + kl];
            a.y = rs[hl * SVP + kk * 4 + kl + 1];
#pragma unroll
            for (int j = 0; j < 3; ++j) {
                const int bt0 = (wave * 3 + j) * 16;
                v2f b;
                b.x = xs[(kk * 4 + kl) * XSP + bt0 + hl];
                b.y = xs[(kk * 4 + kl + 1) * XSP + bt0 + hl];
                acc[j] = wmma_f32(a, b, acc[j]);
            }
        }
    }

    // store Xr[e][v][bt] (coalesced: lanes hold consecutive bt)
#pragma unroll
    for (int j = 0; j < 3; ++j) {
        const int bt = (wave * 3 + j) * 16 + hl;
#pragma unroll
        for (int i = 0; i < 8; ++i) {
            const int m = i + ((lane >> 4) << 3);
            Xr[((size_t)e * VV + v0 + m) * BT + bt] = acc[j][i];
        }
    }
}

// ---------------------------------------------------------------------------
// Kernel 2: per (t, v): y[b,:] = leaky( x_list[b,192] @ W[t,v,192,64] + bias )
// x_list built in LDS (async x_rel/sape copies + VALU time-emb sum).
// W streamed in two 24 KB chunks via async b128 (604 MB total, read once).
// 8 waves each own one 16x16 output tile; 48 f32-WMMA K-steps.
// ---------------------------------------------------------------------------
#define XLP 193                 // padded row of x_list (192 + 1)
#define WCH 96                  // W K-chunk rows
#define WTP 68                  // padded W row (64 + 4)

__global__ __launch_bounds__(256) void k2_connect(
        const float* __restrict__ Xr,
        const int*   __restrict__ n_route,
        const int*   __restrict__ s_week,
        const int*   __restrict__ s_day,
        const float* __restrict__ sape,
        const float* __restrict__ dow,
        const float* __restrict__ tod,
        const float* __restrict__ W,
        const float* __restrict__ bias,
        float* __restrict__ out) {
    const int v    = blockIdx.x;
    const int t    = blockIdx.y;
    const int tid  = threadIdx.x;
    const int lane = tid & 31;
    const int wave = tid >> 5;
    const int hl   = lane & 15;
    const int kl   = (lane >> 4) << 1;

    __shared__ float xl[BB * XLP];           // x_list [32][192]
    __shared__ float wt[WCH * WTP];          // W chunk [96][64]

    // ---- build x_list tile ----
    // f in [0,64): x_rel from Xr[e][v][t][b]  (async b32 gather)
    {
        const float* xrbase = Xr + ((size_t)v * TT + t) * BB;   // uniform
        for (int idx = tid; idx < BB * EE; idx += 256) {
            const int e = idx >> 5, b = idx & 31;
            async_b32(lds_off(&xl[b * XLP + e]),
                      (uint32_t)(e * (VV * TT * BB) + b) * 4u, xrbase);
        }
    }
    // f in [64,128): spatial abs-pos gather (async b128: 64 floats per b)
    for (int idx = tid; idx < BB * (EE / 4); idx += 256) {
        const int b = idx >> 4, q = idx & 15;
        const int n = n_route[b * VV + v];
        async_b128(lds_off(&xl[b * XLP + 64 + q * 4]),
                   (uint32_t)(((n * TT + t) * EE) + q * 4) * 4u, sape);
    }
    // f in [128,192): time embeddings need a VALU add -> regular path
    for (int idx = tid; idx < BB * EE; idx += 256) {
        const int b  = idx >> 6, e = idx & 63;
        const int wk = s_week[b * TT + t];
        const int dy = s_day[b * TT + t];
        xl[b * XLP + 128 + e] = dow[((size_t)wk * VV + v) * EE + e]
                              + tod[((size_t)dy * VV + v) * EE + e];
    }

    const int b0 = (wave >> 2) * 16;         // output row tile (b)
    const int e0 = (wave & 3) * 16;          // output col tile (e)
    v8f acc = {};

    const float* wbase = W + (((size_t)t * VV + v) * 3 * EE) * EE;

    for (int c = 0; c < 2; ++c) {
        __syncthreads();   // previous wt consumed (and x_list readers gated)
        // stream W chunk: contiguous 96*64 floats, async b128 to padded rows
        const float* wsrc = wbase + (size_t)c * WCH * EE;   // uniform
        for (int idx = tid; idx < WCH * (EE / 16) * 4; idx += 256) {
            const int f = idx >> 4;                 // 16 float4 per row
            const int q = idx & 15;
            async_b128(lds_off(&wt[f * WTP + q * 4]),
                       (uint32_t)idx * 16u, wsrc);
        }
        ASYNC_WAIT();      // this wave's async copies (incl. x_list) done
        __syncthreads();

#pragma unroll
        for (int kk = 0; kk < WCH / 4; ++kk) {      // 24 K-steps per chunk
            const int k = kk * 4 + kl;              // k within chunk
            v2f a, b;
            a.x = xl[(b0 + hl) * XLP + c * WCH + k];
            a.y = xl[(b0 + hl) * XLP + c * WCH + k + 1];
            b.x = wt[k * WTP + e0 + hl];
            b.y = wt[(k + 1) * WTP + e0 + hl];
            acc = wmma_f32(a, b, acc);
        }
    }

    // ---- epilogue: bias + LeakyReLU, coalesced store (lanes = consecutive e)
    const int e  = e0 + hl;
    const float bv = bias[((size_t)t * VV + v) * EE + e];
#pragma unroll
    for (int i = 0; i < 8; ++i) {
        const int b = b0 + i + ((lane >> 4) << 3);
        float y = acc[i] + bv;
        y = y > 0.0f ? y : ALPHA * y;
        out[(((size_t)b * TT + t) * VV + v) * EE + e] = y;
    }
}

// ---------------------------------------------------------------------------
// Launcher. Workspace layout: Xt (E*V*T*B floats) | Xr (E*V*T*B floats)
// = 2 * 100.7 MB.
// ---------------------------------------------------------------------------
extern "C" void kernel_launch(void* const* d_in, const int* in_sizes, int n_in,
                              void* d_out, int out_size, void* d_ws, size_t ws_size,
                              hipStream_t stream) {
    const float* inputs  = (const float*)d_in[0];
    const int*   n_route = (const int*)  d_in[1];
    const int*   s_week  = (const int*)  d_in[2];
    const int*   s_day   = (const int*)  d_in[3];
    const float* srpe    = (const float*)d_in[4];
    const float* sape    = (const float*)d_in[5];
    const float* dow     = (const float*)d_in[6];
    const float* tod     = (const float*)d_in[7];
    const float* W       = (const float*)d_in[8];
    const float* bias    = (const float*)d_in[9];
    float* out = (float*)d_out;

    float* Xt = (float*)d_ws;
    float* Xr = Xt + (size_t)EE * VV * BT;   // +25,165,824 floats

    k0_transpose<<<dim3(VV, TT), 256, 0, stream>>>(inputs, Xt);
    k1_relmix  <<<dim3(VV / 16, EE), 256, 0, stream>>>(srpe, Xt, Xr);
    k2_connect <<<dim3(VV, TT), 256, 0, stream>>>(Xr, n_route, s_week, s_day,
                                                  sape, dow, tod, W, bias, out);
}